// FlashMoEModel_58162447122562
// MI455X (gfx1250) — compile-verified
//
#include <hip/hip_runtime.h>
#include <math.h>

// ---------------------------------------------------------------------------
// FlashMoE on MI455X (gfx1250): three fp32 NT-GEMMs via V_WMMA_F32_16X16X4_F32
//   enc = x @ W_enc^T + b                      (mode 0)
//   h   = silu(enc @ Uflat^T) * coef[b, e]     (mode 1, Uflat = U viewed [E*R, D])
//   out = enc + h @ Vflat                      (mode 2, Vflat pre-transposed)
// Data movement: double-buffered GLOBAL_LOAD_ASYNC_TO_LDS_B128 (ASYNCcnt),
// so HBM->LDS for tile t+1 overlaps the WMMA pipeline on tile t.
// Deterministic: no atomics; every ws byte written before read.
// Workspace (~133 MB): enc[64MB] | h[64MB] | Wv[4MB] | coef[1MB]
// ---------------------------------------------------------------------------

#define BTOK   16384
#define DMODEL 1024
#define NEXP   16
#define RANK   64

typedef __attribute__((ext_vector_type(2))) float v2f;
typedef __attribute__((ext_vector_type(8))) float v8f;
typedef __attribute__((ext_vector_type(4))) int   v4i;

#define LDS_STRIDE 36                    // conflict-free: 36*i mod 64 = 4-aligned, distinct
#define TILE_F     (128 * LDS_STRIDE)    // floats per (A or W) tile buffer
#define SMEM_BYTES (4 * TILE_F * 4)      // 2 buffers x (A,W) = 73728 B dynamic LDS

#if __has_builtin(__builtin_amdgcn_global_load_async_to_lds_b128)
#define HAVE_ASYNC_LDS 1
#endif

#pragma clang diagnostic push
#pragma clang diagnostic ignored "-Wold-style-cast"

__device__ __forceinline__ void async_cp16(const float* g, float* l)
{
#if HAVE_ASYNC_LDS
    // (int4 AS(1)* src_global, int4 AS(3)* dst_lds, imm offset, imm cpol)
    __builtin_amdgcn_global_load_async_to_lds_b128(
        (__attribute__((address_space(1))) v4i*)g,
        (__attribute__((address_space(3))) v4i*)l, 0, 0);
#else
    *(float4*)l = *(const float4*)g;
#endif
}

__device__ __forceinline__ void wait_async_all()
{
#if HAVE_ASYNC_LDS
#if __has_builtin(__builtin_amdgcn_s_wait_asynccnt)
    __builtin_amdgcn_s_wait_asynccnt(0);
#else
    asm volatile("s_wait_asynccnt 0x0" ::: "memory");
#endif
#endif
}

// ---------------------------------------------------------------------------
// NT GEMM: Out[m,n] = sum_k A[m,k] * W[n,k]  (+ epilogue)
//   mode 0: + aux[n]                       (bias)
//   mode 1: silu(v) * aux[m*16 + n/64]     (per-token-per-expert coef)
//   mode 2: + aux[m*1024 + n]              (residual = encoded)
// Grid (M/128, N/128); 256 threads = 8 waves in 2(M) x 4(N); wave tile 64x32
// = 4x2 WMMA 16x16 fp32 accumulators. K staged in 32-wide LDS tiles,
// double-buffered via async copies.
// ---------------------------------------------------------------------------
__global__ __launch_bounds__(256) void gemm_nt_wmma(
    const float* __restrict__ A,    // [M,1024]
    const float* __restrict__ W,    // [1024,1024] row-major (NT)
    float* __restrict__ Out,        // [M,1024]
    const float* __restrict__ aux,
    int mode)
{
    const int K = 1024;
    const int NKT = K / 32;                 // 32 K-tiles

    extern __shared__ float smem[];         // [2 bufs][A | W][128*LDS_STRIDE]

    const int tid  = threadIdx.x;
    const int lane = tid & 31;
    const int wave = tid >> 5;
    const int wm   = wave >> 2;             // 0..1  (M)
    const int wn   = wave & 3;              // 0..3  (N)
    const int l16  = lane & 15;
    const int hh   = lane >> 4;             // half-wave: K pair select / M+8
    const int m0   = blockIdx.x * 128;
    const int n0   = blockIdx.y * 128;

    v8f acc[4][2];
#pragma unroll
    for (int i = 0; i < 4; ++i)
#pragma unroll
        for (int j = 0; j < 2; ++j) acc[i][j] = (v8f){0,0,0,0,0,0,0,0};

    const int lrow = tid >> 3;              // 0..31
    const int lc4  = tid & 7;               // 0..7 (float4 column)

#if HAVE_ASYNC_LDS
    // ---- issue one 128x32 A-tile + W-tile into LDS buffer `b` (8 async b128/thread)
    auto issue_tile = [&](int b, int k0) {
        float* Asb = smem + b * 2 * TILE_F;
        float* Wsb = Asb + TILE_F;
#pragma unroll
        for (int it = 0; it < 4; ++it) {
            const int rr = lrow + 32 * it;
            async_cp16(A + (size_t)(m0 + rr) * K + k0 + lc4 * 4,
                       Asb + rr * LDS_STRIDE + lc4 * 4);
            async_cp16(W + (size_t)(n0 + rr) * K + k0 + lc4 * 4,
                       Wsb + rr * LDS_STRIDE + lc4 * 4);
        }
    };

    issue_tile(0, 0);                       // prologue: tile 0 in flight

    for (int kt = 0; kt < NKT; ++kt) {
        wait_async_all();                   // this wave's tile-kt copies landed
        __syncthreads();                    // all waves' copies visible; buffer
                                            // (kt+1)&1 fully consumed by everyone
        if (kt + 1 < NKT)
            issue_tile((kt + 1) & 1, (kt + 1) * 32);   // overlap with compute
        if (kt + 2 < NKT) {                 // warm L2 two tiles ahead
            __builtin_prefetch(A + (size_t)(m0 + lrow) * K + (kt + 2) * 32, 0, 1);
            __builtin_prefetch(W + (size_t)(n0 + lrow) * K + (kt + 2) * 32, 0, 1);
        }
        const float* As = smem + (kt & 1) * 2 * TILE_F;
        const float* Ws = As + TILE_F;
#pragma unroll
        for (int ks = 0; ks < 8; ++ks) {
            const int kk = ks * 4 + 2 * hh;
            v2f afr[4], bfr[2];
#pragma unroll
            for (int i = 0; i < 4; ++i)
                afr[i] = *(const v2f*)(As + (wm * 64 + i * 16 + l16) * LDS_STRIDE + kk);
#pragma unroll
            for (int j = 0; j < 2; ++j)
                bfr[j] = *(const v2f*)(Ws + (wn * 32 + j * 16 + l16) * LDS_STRIDE + kk);
#pragma unroll
            for (int i = 0; i < 4; ++i)
#pragma unroll
                for (int j = 0; j < 2; ++j)
                    acc[i][j] = __builtin_amdgcn_wmma_f32_16x16x4_f32(
                        false, afr[i], false, bfr[j],
                        (short)0, acc[i][j], false, false);
        }
    }
#else
    // ---- fallback: single-buffered synchronous staging
    float* As = smem;
    float* Ws = smem + TILE_F;
    for (int kt = 0; kt < NKT; ++kt) {
        const int k0 = kt * 32;
        __syncthreads();
        float4 av[4], wv[4];
#pragma unroll
        for (int it = 0; it < 4; ++it) {
            const int rr = lrow + 32 * it;
            av[it] = *(const float4*)(A + (size_t)(m0 + rr) * K + k0 + lc4 * 4);
            wv[it] = *(const float4*)(W + (size_t)(n0 + rr) * K + k0 + lc4 * 4);
        }
#pragma unroll
        for (int it = 0; it < 4; ++it) {
            const int rr = lrow + 32 * it;
            *(float4*)(As + rr * LDS_STRIDE + lc4 * 4) = av[it];
            *(float4*)(Ws + rr * LDS_STRIDE + lc4 * 4) = wv[it];
        }
        __syncthreads();
#pragma unroll
        for (int ks = 0; ks < 8; ++ks) {
            const int kk = ks * 4 + 2 * hh;
            v2f afr[4], bfr[2];
#pragma unroll
            for (int i = 0; i < 4; ++i)
                afr[i] = *(const v2f*)(As + (wm * 64 + i * 16 + l16) * LDS_STRIDE + kk);
#pragma unroll
            for (int j = 0; j < 2; ++j)
                bfr[j] = *(const v2f*)(Ws + (wn * 32 + j * 16 + l16) * LDS_STRIDE + kk);
#pragma unroll
            for (int i = 0; i < 4; ++i)
#pragma unroll
                for (int j = 0; j < 2; ++j)
                    acc[i][j] = __builtin_amdgcn_wmma_f32_16x16x4_f32(
                        false, afr[i], false, bfr[j],
                        (short)0, acc[i][j], false, false);
        }
    }
#endif

    // Epilogue + store. C/D layout: VGPR r -> M = r + 8*hh, N = lane&15.
#pragma unroll
    for (int i = 0; i < 4; ++i) {
#pragma unroll
        for (int j = 0; j < 2; ++j) {
            const int col = n0 + wn * 32 + j * 16 + l16;
#pragma unroll
            for (int r = 0; r < 8; ++r) {
                const int row = m0 + wm * 64 + i * 16 + 8 * hh + r;
                float v = acc[i][j][r];
                if (mode == 0) {
                    v += aux[col];
                } else if (mode == 1) {
                    const float s = v / (1.0f + __expf(-v));   // silu
                    v = s * aux[row * NEXP + (col >> 6)];      // col/64 = expert
                } else {
                    v += aux[(size_t)row * DMODEL + col];      // residual
                }
                Out[(size_t)row * DMODEL + col] = v;
            }
        }
    }
}

#pragma clang diagnostic pop

// ---------------------------------------------------------------------------
// Pre-transpose V [E,D,R] -> Wv [D, E*R] row-major (NT layout for GEMM3):
//   Wv[d*1024 + e*64 + r] = V[e*D*R + d*R + r]
// ---------------------------------------------------------------------------
__global__ __launch_bounds__(256) void transpose_v(
    const float* __restrict__ V, float* __restrict__ Wv)
{
    const int idx = blockIdx.x * 256 + threadIdx.x;     // 0..2^20-1
    const int d = idx >> 10;
    const int k = idx & 1023;
    const int e = k >> 6;
    const int r = k & 63;
    Wv[idx] = V[((size_t)e << 16) + (d << 6) + r];      // e*D*R + d*R + r
}

// ---------------------------------------------------------------------------
// Gating: one wave per token. logits_e = enc[b,:] . W_gate[e,:], tau = 1.
// top-2 (first-index-wins, == top-2 of top-4), softmax over the 2 selected,
// coef[b,e] = w * gamma[e] (dense, zeros elsewhere).
// ---------------------------------------------------------------------------
__global__ __launch_bounds__(256) void gate_kernel(
    const float* __restrict__ enc, const float* __restrict__ Wg,
    const float* __restrict__ gamma, float* __restrict__ coef)
{
    const int lane = threadIdx.x & 31;
    const int wave = threadIdx.x >> 5;
    const int b    = blockIdx.x * 8 + wave;

    const float4* xr = (const float4*)(enc + (size_t)b * DMODEL);
    float acc[NEXP];
#pragma unroll
    for (int e = 0; e < NEXP; ++e) acc[e] = 0.0f;

#pragma unroll
    for (int it = 0; it < 8; ++it) {
        const int c4 = it * 32 + lane;                  // float4 index 0..255
        const float4 xv = xr[c4];
#pragma unroll
        for (int e = 0; e < NEXP; ++e) {
            const float4 wv = ((const float4*)(Wg + e * DMODEL))[c4];
            acc[e] += xv.x * wv.x + xv.y * wv.y + xv.z * wv.z + xv.w * wv.w;
        }
    }
#pragma unroll
    for (int off = 16; off > 0; off >>= 1)              // wave32 butterfly
#pragma unroll
        for (int e = 0; e < NEXP; ++e)
            acc[e] += __shfl_xor(acc[e], off, 32);

    if (lane == 0) {
        int i1 = 0; float v1 = acc[0];
        for (int e = 1; e < NEXP; ++e)
            if (acc[e] > v1) { v1 = acc[e]; i1 = e; }
        int i2 = -1; float v2 = -INFINITY;
        for (int e = 0; e < NEXP; ++e)
            if (e != i1 && acc[e] > v2) { v2 = acc[e]; i2 = e; }
        const float e2 = __expf(v2 - v1);               // stable softmax, max=v1
        const float s  = 1.0f + e2 + 1e-12f;
        const float w1 = 1.0f / s, w2 = e2 / s;
        for (int e = 0; e < NEXP; ++e) {
            float c = 0.0f;
            if (e == i1) c = w1;
            else if (e == i2) c = w2;
            coef[b * NEXP + e] = c * gamma[e];
        }
    }
}

// ---------------------------------------------------------------------------
extern "C" void kernel_launch(void* const* d_in, const int* in_sizes, int n_in,
                              void* d_out, int out_size, void* d_ws, size_t ws_size,
                              hipStream_t stream)
{
    const float* x      = (const float*)d_in[0];
    const float* W_enc  = (const float*)d_in[1];
    const float* b_enc  = (const float*)d_in[2];
    const float* W_gate = (const float*)d_in[3];
    const float* U      = (const float*)d_in[4];   // [E,R,D] == Uflat [1024,1024] NT
    const float* V      = (const float*)d_in[5];   // [E,D,R]
    const float* gamma  = (const float*)d_in[6];
    float* out = (float*)d_out;

    char* ws = (char*)d_ws;
    float* enc  = (float*)(ws);                          // 64 MB
    float* h    = (float*)(ws + (67108864ull));          // 64 MB
    float* Wv   = (float*)(ws + (134217728ull));         // 4 MB
    float* coef = (float*)(ws + (138412032ull));         // 1 MB

    // dynamic LDS (72 KB > default 64 KB cap; gfx1250 WGP has 320 KB)
    (void)hipFuncSetAttribute((const void*)gemm_nt_wmma,
                              hipFuncAttributeMaxDynamicSharedMemorySize, SMEM_BYTES);

    transpose_v<<<dim3(4096), dim3(256), 0, stream>>>(V, Wv);

    dim3 g(BTOK / 128, DMODEL / 128);   // (128, 8)
    dim3 blk(256);

    // enc = x @ W_enc^T + b
    gemm_nt_wmma<<<g, blk, SMEM_BYTES, stream>>>(x, W_enc, enc, b_enc, 0);
    // coef[b,e] = topk softmax weight * gamma
    gate_kernel<<<dim3(BTOK / 8), blk, 0, stream>>>(enc, W_gate, gamma, coef);
    // h = silu(enc @ Uflat^T) * coef
    gemm_nt_wmma<<<g, blk, SMEM_BYTES, stream>>>(enc, U, h, coef, 1);
    // out = enc + h @ Vflat
    gemm_nt_wmma<<<g, blk, SMEM_BYTES, stream>>>(h, Wv, out, enc, 2);
}